// CrossAttention_32031866094253
// MI455X (gfx1250) — compile-verified
//
#include <hip/hip_runtime.h>
#include <hip/hip_bf16.h>
#include <math.h>

typedef __attribute__((ext_vector_type(2))) float v2f;
typedef __attribute__((ext_vector_type(8))) float v8f;

constexpr int kB  = 4;
constexpr int kH  = 8;
constexpr int kHD = 32;
constexpr int kC  = 256;     // model dim (== D)
constexpr int kNq = 2048;
constexpr int kNs = 2048;

// padded strides to avoid LDS bank conflicts (64 banks x 4B)
constexpr int kAStr = 260;   // 260 % 64 == 4 -> 16 lanes hit 16 distinct banks
constexpr int kSStr = 2052;  // 2052 % 64 == 4

__device__ __forceinline__ v8f wmma_f32(v2f a, v2f b, v8f c) {
  return __builtin_amdgcn_wmma_f32_16x16x4_f32(false, a, false, b, (short)0, c,
                                               false, false);
}

// ---------------------------------------------------------------------------
// Tensor Data Mover staging (CDNA5): 2D tile from global -> LDS, with the
// TDM pad feature reproducing the bank-conflict row padding.
// This toolchain exposes the 6-arg builtin:
//   (u32x4 g0, i32x8 g1, i32x4 g2, i32x4 g3, i32x8 g4, i32 cpol)
// Guarded with __has_builtin; the host parse / fallback path uses a loop.
// ---------------------------------------------------------------------------
#if defined(__AMDGCN__) && __has_builtin(__builtin_amdgcn_tensor_load_to_lds) && \
    __has_builtin(__builtin_amdgcn_s_wait_tensorcnt)
#define USE_TDM 1
#else
#define USE_TDM 0
#endif

#if USE_TDM
typedef __attribute__((ext_vector_type(4))) unsigned int u32x4;
typedef __attribute__((ext_vector_type(8))) int i32x8;
typedef __attribute__((ext_vector_type(4))) int i32x4;

// D# group layouts per CDNA5 ISA ch.8 (CDNA5 ISA).
__device__ __forceinline__ void tdm_load_tile_2d(
    const float* gsrc, const float* ldsDst, unsigned tileD0, unsigned tileD1,
    unsigned tensorD0, unsigned tensorD1, unsigned rowStrideElems,
    unsigned padInterval, unsigned padAmountM1, bool padEnable) {
  const unsigned long long ga = (unsigned long long)(size_t)gsrc;
  const unsigned lds = (unsigned)(size_t)ldsDst;  // low 32 bits = LDS offset
  u32x4 g0;
  g0[0] = 1u;                                   // count=1 (valid), user load
  g0[1] = lds;                                  // lds_addr
  g0[2] = (unsigned)ga;                         // global_addr[31:0]
  g0[3] = (unsigned)((ga >> 32) & 0x01FFFFFFu)  // global_addr[56:32]
          | (2u << 30);                         // type = 2 ("image")
  i32x8 g1;
  g1[0] = (int)((2u << 16)                              // data_size = 4 bytes
                | (padEnable ? (1u << 20) : 0u)         // pad_enable
                | (padInterval << 22)                   // pad every 2^(k+1) dw
                | (padAmountM1 << 25));                 // pad (v+1) dwords
  g1[1] = (int)((tensorD0 & 0xFFFFu) << 16);            // tensor_dim0[15:0]
  g1[2] = (int)((tensorD0 >> 16) | ((tensorD1 & 0xFFFFu) << 16));
  g1[3] = (int)((tensorD1 >> 16) | (tileD0 << 16));     // tile_dim0
  g1[4] = (int)(tileD1 & 0xFFFFu);                      // tile_dim1 (dim2=0)
  g1[5] = (int)rowStrideElems;                          // tensor_dim0_stride
  g1[6] = 0;
  g1[7] = 0;
  i32x4 z4 = {};
  i32x8 z8 = {};
  __builtin_amdgcn_tensor_load_to_lds(g0, g1, z4, z4, z8, 0);
}
#endif

// ---------------------------------------------------------------------------
// Kernel 1: y = x @ W^T  for q (blockIdx.y==0) and k (blockIdx.y==1)
// x: (B*N, 256), W: (256,256) row-major, y[m,n] = sum_k x[m,k]*W[n,k]
// grid.x = (B*N)/16, block = 256 (8 waves); each wave owns 32 output cols.
// ---------------------------------------------------------------------------
__global__ __launch_bounds__(256) void proj_kernel(
    const float* __restrict__ q, const float* __restrict__ k,
    const float* __restrict__ W, float* __restrict__ qh,
    float* __restrict__ kh) {
  __shared__ float Ash[16 * kAStr];
  const float* x = (blockIdx.y == 0) ? q : k;
  float* y       = (blockIdx.y == 0) ? qh : kh;
  const int m0g = blockIdx.x * 16;

#if USE_TDM
  if (threadIdx.x < 32) {
    // 16 rows x 256 cols; TDM pad: +4 dwords after each 256-dword row -> 260
    tdm_load_tile_2d(x + (size_t)m0g * kC, Ash, kC, 16, kC,
                     (unsigned)(kB * kNq), kC, 7u, 3u, true);
    __builtin_amdgcn_s_wait_tensorcnt(0);
  }
#else
  for (int i = threadIdx.x; i < 16 * kC; i += 256) {
    int r = i >> 8, c = i & 255;
    Ash[r * kAStr + c] = x[(size_t)(m0g + r) * kC + c];
  }
#endif
  __syncthreads();

  const int wave = threadIdx.x >> 5;
  const int lane = threadIdx.x & 31;
  const int lm = lane & 15, hi = lane >> 4;

  for (int nt = 0; nt < 2; ++nt) {
    const int n0 = wave * 32 + nt * 16;
    const int n  = n0 + lm;
    v8f acc = {};
    const float* wrow = W + (size_t)n * kC;
    for (int k0 = 0; k0 < kC; k0 += 4) {
      v2f a, b;
      a.x = Ash[lm * kAStr + k0 + 2 * hi];
      a.y = Ash[lm * kAStr + k0 + 2 * hi + 1];
      b.x = wrow[k0 + 2 * hi];
      b.y = wrow[k0 + 2 * hi + 1];
      acc = wmma_f32(a, b, acc);
    }
    for (int r = 0; r < 8; ++r) {
      int m = r + 8 * hi;
      y[(size_t)(m0g + m) * kC + n] = acc[r];
    }
  }
}

// ---------------------------------------------------------------------------
// Kernel 2: attention. One workgroup per (b, h, 32-row q tile).
// Uses ~264.5 KiB LDS (CDNA5 WGP has 320 KB) to hold the full 32 x 2048 score
// tile so attn is written once and P@V reads it back from LDS.
// ---------------------------------------------------------------------------
__global__ __launch_bounds__(256) void attn_kernel(
    const float* __restrict__ qh, const float* __restrict__ kh,
    const float* __restrict__ vin, const int* __restrict__ smask,
    float* __restrict__ attn, float* __restrict__ obuf) {
  extern __shared__ float smem[];
  float* S   = smem;                    // 32 * kSStr
  float* qsh = S + 32 * kSStr;          // 32 * 32
  float* Osh = qsh + 32 * 32;           // 32 * 32

  const int qt = blockIdx.x & 63;
  const int h  = (blockIdx.x >> 6) & 7;
  const int b  = blockIdx.x >> 9;
  const int q0 = qt * 32;
  const int tid = threadIdx.x;

  for (int i = tid; i < 32 * 32; i += 256) Osh[i] = 0.0f;
#if USE_TDM
  if (tid < 32) {
    // 32 rows x 32 cols tile of qh at (b*Nq+q0, h*32); row stride 256, no pad
    tdm_load_tile_2d(qh + ((size_t)b * kNq + q0) * kC + h * kHD, qsh, kHD, 32,
                     kC, (unsigned)(kB * kNq), kC, 0u, 0u, false);
    __builtin_amdgcn_s_wait_tensorcnt(0);
  }
#else
  for (int i = tid; i < 32 * 32; i += 256) {
    int r = i >> 5, c = i & 31;
    qsh[i] = qh[((size_t)b * kNq + q0 + r) * kC + h * kHD + c];
  }
#endif
  __syncthreads();

  const int wave = tid >> 5, lane = tid & 31;
  const int lm = lane & 15, hi = lane >> 4;
  const float temp = 0.17677669529663687f;  // 1/sqrt(32)

  // ---- Phase 1: S = (qh_tile @ kh^T) * temp + (-1000 * mask) ----
  const size_t kbase = (size_t)b * kNs * kC + (size_t)h * kHD;
  for (int mt = 0; mt < 2; ++mt) {
    for (int nt = 0; nt < 16; ++nt) {
      const int n0 = wave * 256 + nt * 16;
      const int n  = n0 + lm;
      const float* krow = kh + kbase + (size_t)n * kC;
      v8f acc = {};
      for (int kk = 0; kk < 8; ++kk) {
        int k0 = kk * 4;
        v2f a, bb;
        a.x  = qsh[(mt * 16 + lm) * 32 + k0 + 2 * hi];
        a.y  = qsh[(mt * 16 + lm) * 32 + k0 + 2 * hi + 1];
        bb.x = krow[k0 + 2 * hi];
        bb.y = krow[k0 + 2 * hi + 1];
        acc = wmma_f32(a, bb, acc);
      }
      const float mval = -1000.0f * (float)smask[(size_t)b * kNs + n];
      for (int r = 0; r < 8; ++r) {
        int m = mt * 16 + r + 8 * hi;
        S[m * kSStr + n] = acc[r] * temp + mval;
      }
    }
  }
  __syncthreads();

  // ---- Phase 2: row softmax; write attn; keep normalized P in LDS ----
  for (int rr = 0; rr < 4; ++rr) {
    const int m = wave * 4 + rr;
    float mx = -INFINITY;
    for (int c = lane; c < kNs; c += 32) mx = fmaxf(mx, S[m * kSStr + c]);
    for (int o = 16; o > 0; o >>= 1) mx = fmaxf(mx, __shfl_xor(mx, o, 32));
    float sum = 0.0f;
    for (int c = lane; c < kNs; c += 32) {
      float e = expf(S[m * kSStr + c] - mx);
      S[m * kSStr + c] = e;
      sum += e;
    }
    for (int o = 16; o > 0; o >>= 1) sum += __shfl_xor(sum, o, 32);
    const float inv = 1.0f / sum;
    float* arow = attn + (((size_t)(b * kH + h) * kNq) + q0 + m) * kNs;
    for (int c = lane; c < kNs; c += 32) {
      float p = S[m * kSStr + c] * inv;
      S[m * kSStr + c] = p;
      arow[c] = p;
    }
  }
  __syncthreads();

  // ---- Phase 3: O = P @ vh ; waves split K=2048, merge via LDS atomics ----
  const size_t vbase = (size_t)b * kNs * kC + (size_t)h * kHD;
  for (int mt = 0; mt < 2; ++mt) {
    for (int nt = 0; nt < 2; ++nt) {
      v8f acc = {};
      for (int kk = 0; kk < 64; ++kk) {
        int k0 = wave * 256 + kk * 4;
        v2f a, bb;
        a.x  = S[(mt * 16 + lm) * kSStr + k0 + 2 * hi];
        a.y  = S[(mt * 16 + lm) * kSStr + k0 + 2 * hi + 1];
        bb.x = vin[vbase + (size_t)(k0 + 2 * hi) * kC + nt * 16 + lm];
        bb.y = vin[vbase + (size_t)(k0 + 2 * hi + 1) * kC + nt * 16 + lm];
        acc = wmma_f32(a, bb, acc);
      }
      for (int r = 0; r < 8; ++r) {
        int m = mt * 16 + r + 8 * hi;
        atomicAdd(&Osh[m * 32 + nt * 16 + lm], acc[r]);
      }
    }
  }
  __syncthreads();
  for (int i = tid; i < 32 * 32; i += 256) {
    int r = i >> 5, c = i & 31;
    obuf[((size_t)b * kNq + q0 + r) * kC + h * kHD + c] = Osh[i];
  }
}

// ---------------------------------------------------------------------------
// Kernel 3: x = obuf @ W_fc^T + b_fc + idt ; out = LayerNorm(x)*w + b
// grid.x = (B*Nq)/16 ; block = 256 (8 waves)
// ---------------------------------------------------------------------------
__global__ __launch_bounds__(256) void fcln_kernel(
    const float* __restrict__ obuf, const float* __restrict__ Wfc,
    const float* __restrict__ bfc, const float* __restrict__ idt,
    const float* __restrict__ lnw, const float* __restrict__ lnb,
    float* __restrict__ out) {
  __shared__ float Ash[16 * kAStr];
  __shared__ float Xsh[16 * kAStr];
  const int m0g = blockIdx.x * 16;

#if USE_TDM
  if (threadIdx.x < 32) {
    tdm_load_tile_2d(obuf + (size_t)m0g * kC, Ash, kC, 16, kC,
                     (unsigned)(kB * kNq), kC, 7u, 3u, true);
    __builtin_amdgcn_s_wait_tensorcnt(0);
  }
#else
  for (int i = threadIdx.x; i < 16 * kC; i += 256) {
    int r = i >> 8, c = i & 255;
    Ash[r * kAStr + c] = obuf[(size_t)(m0g + r) * kC + c];
  }
#endif
  __syncthreads();

  const int wave = threadIdx.x >> 5, lane = threadIdx.x & 31;
  const int lm = lane & 15, hi = lane >> 4;

  for (int nt = 0; nt < 2; ++nt) {
    const int n0 = wave * 32 + nt * 16;
    const int n  = n0 + lm;
    const float* wrow = Wfc + (size_t)n * kC;
    v8f acc = {};
    for (int k0 = 0; k0 < kC; k0 += 4) {
      v2f a, bb;
      a.x  = Ash[lm * kAStr + k0 + 2 * hi];
      a.y  = Ash[lm * kAStr + k0 + 2 * hi + 1];
      bb.x = wrow[k0 + 2 * hi];
      bb.y = wrow[k0 + 2 * hi + 1];
      acc = wmma_f32(a, bb, acc);
    }
    const float bias = bfc[n];
    for (int r = 0; r < 8; ++r) {
      int m = r + 8 * hi;
      Xsh[m * kAStr + n] = acc[r] + bias + idt[(size_t)(m0g + m) * kC + n];
    }
  }
  __syncthreads();

  for (int rr = 0; rr < 2; ++rr) {
    const int m = wave * 2 + rr;
    float s = 0.0f;
    for (int c = lane; c < kC; c += 32) s += Xsh[m * kAStr + c];
    for (int o = 16; o > 0; o >>= 1) s += __shfl_xor(s, o, 32);
    const float mu = s * (1.0f / kC);
    float vs = 0.0f;
    for (int c = lane; c < kC; c += 32) {
      float d = Xsh[m * kAStr + c] - mu;
      vs += d * d;
    }
    for (int o = 16; o > 0; o >>= 1) vs += __shfl_xor(vs, o, 32);
    const float inv = rsqrtf(vs * (1.0f / kC) + 1e-5f);
    for (int c = lane; c < kC; c += 32) {
      out[(size_t)(m0g + m) * kC + c] =
          (Xsh[m * kAStr + c] - mu) * inv * lnw[c] + lnb[c];
    }
  }
}

// ---------------------------------------------------------------------------
extern "C" void kernel_launch(void* const* d_in, const int* in_sizes, int n_in,
                              void* d_out, int out_size, void* d_ws,
                              size_t ws_size, hipStream_t stream) {
  const float* k    = (const float*)d_in[0];
  const float* v    = (const float*)d_in[1];
  const float* q    = (const float*)d_in[2];
  const float* idt  = (const float*)d_in[3];
  const int*   msk  = (const int*)d_in[4];
  const float* Wqk  = (const float*)d_in[5];
  const float* Wfc  = (const float*)d_in[6];
  const float* bfc  = (const float*)d_in[7];
  const float* lnw  = (const float*)d_in[8];
  const float* lnb  = (const float*)d_in[9];

  const size_t rows = (size_t)kB * kNq;          // 8192
  float* qh   = (float*)d_ws;                    // rows*256 f32
  float* kh   = qh + rows * kC;                  // rows*256 f32
  float* obuf = kh + rows * kC;                  // rows*256 f32

  float* out_ln = (float*)d_out;
  float* attn   = out_ln + rows * kC;            // B*H x Nq x Ns

  // 1) shared q/k projection
  proj_kernel<<<dim3(rows / 16, 2), 256, 0, stream>>>(q, k, Wqk, qh, kh);

  // 2) attention (scores + softmax + attn output + P@V)
  const size_t smem = (size_t)(32 * kSStr + 32 * 32 + 32 * 32) * sizeof(float);
  attn_kernel<<<dim3(kB * kH * (kNq / 32)), 256, smem, stream>>>(
      qh, kh, v, msk, attn, obuf);

  // 3) final FC + residual + LayerNorm
  fcln_kernel<<<dim3(rows / 16), 256, 0, stream>>>(obuf, Wfc, bfc, idt, lnw,
                                                   lnb, out_ln);
}